// TransferringAttentionLayer_87952340287890
// MI455X (gfx1250) — compile-verified
//
#include <hip/hip_runtime.h>

// Problem constants (from reference): B=4,P=6,Q=4,N=512,NHID=32
#define NN     512                 // N
#define WD_    128                 // 4*NHID
#define NBATCH 96                  // B*P*Q
#define NROWS  (NBATCH * NN)       // 49152
#define ALPHA_ 0.2f

typedef __attribute__((ext_vector_type(2))) float v2f;
typedef __attribute__((ext_vector_type(8))) float v8f;

// ---------------------------------------------------------------------------
// Kernel 1: wa1[j] = dot(W[j,:], a[0:128]); wa2[j] = dot(W[j,:], a[128:256])
// (folds the 128x128 weight into two 128-vectors; kills the big GEMM)
// ---------------------------------------------------------------------------
__global__ void wa_kernel(const float* __restrict__ W,
                          const float* __restrict__ a,
                          float* __restrict__ wa) {
    int j = threadIdx.x;                     // 0..127
    const float* row = W + j * WD_;
    float acc1 = 0.f, acc2 = 0.f;
    #pragma unroll 8
    for (int k = 0; k < WD_; ++k) {
        float w = row[k];
        acc1 = fmaf(w, a[k],        acc1);
        acc2 = fmaf(w, a[WD_ + k],  acc2);
    }
    wa[j]       = acc1;   // wa1
    wa[WD_ + j] = acc2;   // wa2
}

// ---------------------------------------------------------------------------
// Kernel 2: S[r, 0:2] = st_feat[r, :] @ [wa1 | wa2]  via V_WMMA_F32_16X16X4_F32
// One wave per 16-row tile; K=128 consumed 4 at a time. B zero-padded to 16
// columns via a branch-free multiplicative mask (loads are unconditional from
// a clamped, always-valid address) so the K-loop has no EXEC-mask churn.
// A layout (16x4 f32): lanes 0-15 -> M=lane, VGPR0=K0, VGPR1=K1;
//                      lanes 16-31 -> M=lane-16, VGPR0=K2, VGPR1=K3.
// B layout (4x16 f32): mirror of A with N=lane&15.
// D layout (16x16 f32): VGPR r -> (M=r, N=lane) lanes 0-15; (M=8+r, N=lane-16).
// ---------------------------------------------------------------------------
__global__ void score_kernel(const float* __restrict__ F,    // [NROWS][WD_]
                             const float* __restrict__ wa,   // [2][WD_]
                             float* __restrict__ s1,
                             float* __restrict__ s2) {
    const int lane  = threadIdx.x & 31;
    const int wave  = (blockIdx.x * blockDim.x + threadIdx.x) >> 5;
    const int row0  = wave * 16;
    if (row0 >= NROWS) return;               // wave-uniform; EXEC stays all-1s

    const int n     = lane & 15;             // tile row (A) / tile col (B)
    const int khalf = (lane >> 4) * 2;       // 0 for lanes 0-15, 2 for 16-31

    const float* frow  = F + (size_t)(row0 + n) * WD_ + khalf;
    const float* brow  = wa + (n < 2 ? n : 1) * WD_ + khalf;  // clamped: always valid
    const float  bmask = (n < 2) ? 1.0f : 0.0f;               // zero-pad cols 2..15

    v8f c = {};
    #pragma unroll
    for (int k0 = 0; k0 < WD_; k0 += 4) {
        v2f A, Bv;
        A.x  = frow[k0];
        A.y  = frow[k0 + 1];
        Bv.x = brow[k0]     * bmask;          // unconditional b64 load + VALU mask
        Bv.y = brow[k0 + 1] * bmask;
        // 8 args: (neg_a, A, neg_b, B, c_mod, C, reuse_a, reuse_b)
        c = __builtin_amdgcn_wmma_f32_16x16x4_f32(
                false, A, false, Bv, (short)0, c, false, false);
    }

    // Column 0 -> s1, column 1 -> s2. Lanes 0..15 hold M=vgpr, 16..31 hold M=8+vgpr.
    const int mbase = row0 + ((lane >> 4) << 3);
    if (n == 0) {
        #pragma unroll
        for (int r = 0; r < 8; ++r) s1[mbase + r] = c[r];
    } else if (n == 1) {
        #pragma unroll
        for (int r = 0; r < 8; ++r) s2[mbase + r] = c[r];
    }
}

// ---------------------------------------------------------------------------
// Kernel 3: per-row softmax of leaky(s1[i]+s2[j]) fused with demand scale.
// One 256-thread block (8 waves) per global row, 2 columns per thread.
// Reductions: wave32 shfl_xor butterfly (no barriers) + 8-entry LDS combine
// (2 barriers total). This kernel moves ~200 MB and is pure HBM-bound.
// ---------------------------------------------------------------------------
__global__ void softmax_kernel(const float* __restrict__ demand,
                               const float* __restrict__ s1,
                               const float* __restrict__ s2,
                               float* __restrict__ out) {
    __shared__ float wmax[8];
    __shared__ float wsum[8];
    const int grow  = blockIdx.x;            // 0..NROWS-1
    const int batch = grow >> 9;             // / NN
    const float* s2b = s2 + (batch << 9);
    const float  s1v = s1[grow];
    const int t    = threadIdx.x;
    const int lane = t & 31;
    const int wid  = t >> 5;

    float e0 = s1v + s2b[t];
    float e1 = s1v + s2b[t + 256];
    e0 = e0 > 0.f ? e0 : ALPHA_ * e0;
    e1 = e1 > 0.f ? e1 : ALPHA_ * e1;

    // ---- row max: wave butterfly, then cross-wave via LDS ----
    float m = fmaxf(e0, e1);
    #pragma unroll
    for (int i = 16; i > 0; i >>= 1) m = fmaxf(m, __shfl_xor(m, i, 32));
    if (lane == 0) wmax[wid] = m;
    __syncthreads();
    float rowmax = wmax[0];
    #pragma unroll
    for (int w = 1; w < 8; ++w) rowmax = fmaxf(rowmax, wmax[w]);

    const float x0 = __expf(e0 - rowmax);
    const float x1 = __expf(e1 - rowmax);

    // ---- row sum: wave butterfly, then cross-wave via LDS ----
    float s = x0 + x1;
    #pragma unroll
    for (int i = 16; i > 0; i >>= 1) s += __shfl_xor(s, i, 32);
    if (lane == 0) wsum[wid] = s;
    __syncthreads();
    float rowsum = wsum[0];
    #pragma unroll
    for (int w = 1; w < 8; ++w) rowsum += wsum[w];
    const float inv = 1.0f / rowsum;

    const float* drow = demand + (size_t)grow * NN;
    float*       orow = out    + (size_t)grow * NN;

    float o0 = drow[t]       * (x0 * inv);
    float o1 = drow[t + 256] * (x1 * inv);
    o0 = o0 > 0.f ? o0 : ALPHA_ * o0;
    o1 = o1 > 0.f ? o1 : ALPHA_ * o1;
    orow[t]       = o0;
    orow[t + 256] = o1;
}

// ---------------------------------------------------------------------------
extern "C" void kernel_launch(void* const* d_in, const int* in_sizes, int n_in,
                              void* d_out, int out_size, void* d_ws, size_t ws_size,
                              hipStream_t stream) {
    const float* demand  = (const float*)d_in[0];   // [B,P,Q,N,N]
    const float* st_feat = (const float*)d_in[1];   // [B,P,Q,N,WD]
    const float* W       = (const float*)d_in[2];   // [WD,WD]
    const float* a       = (const float*)d_in[3];   // [2*WD,1]
    float* out = (float*)d_out;

    float* wa = (float*)d_ws;          // 256 floats: wa1 | wa2
    float* s1 = wa + 2 * WD_;          // NROWS floats
    float* s2 = s1 + NROWS;            // NROWS floats

    wa_kernel<<<1, WD_, 0, stream>>>(W, a, wa);

    // NROWS/16 = 3072 waves; 8 waves (256 threads) per block -> 384 blocks exact
    score_kernel<<<NROWS / 16 / 8, 256, 0, stream>>>(st_feat, wa, s1, s2);

    softmax_kernel<<<NROWS, 256, 0, stream>>>(demand, s1, s2, out);
}